// SkewGaussian_70342974374197
// MI455X (gfx1250) — compile-verified
//
#include <hip/hip_runtime.h>
#include <hip/hip_bf16.h>
#include <math.h>

typedef __attribute__((ext_vector_type(2))) float v2f;
typedef __attribute__((ext_vector_type(8))) float v8f;

#define B_SZ   8
#define N_SZ   4096
#define D_SZ   1024
#define H_SZ   256
#define R_SZ   16
#define TILE_N 16
#define CHUNK  512
#define PITCH  516   // 512 + 4 pad -> bank = (4n + d) % 64, conflict-free

// ---------------------------------------------------------------------------
// h = relu(q @ W1 + b1)   : one block per batch
// ---------------------------------------------------------------------------
__global__ __launch_bounds__(256) void backbone_kernel(
    const float* __restrict__ q, const float* __restrict__ W1,
    const float* __restrict__ b1, float* __restrict__ h) {
  __shared__ float qs[D_SZ];
  const int b = blockIdx.x;
  const int t = threadIdx.x;
  for (int i = t; i < D_SZ; i += 256) qs[i] = q[b * D_SZ + i];
  __syncthreads();
  float acc = b1[t];
  for (int d = 0; d < D_SZ; ++d) acc = fmaf(qs[d], W1[d * H_SZ + t], acc);
  h[b * H_SZ + t] = fmaxf(acc, 0.0f);
}

// ---------------------------------------------------------------------------
// generic head: out = h @ W + bias, with per-mode epilogue
// mode 0: log_d (clamp) + d_inv ; mode 1: mu = q + 0.1*x ; mode 2: alpha ;
// mode 3: L = 0.1*x
// ---------------------------------------------------------------------------
__global__ __launch_bounds__(256) void head_kernel(
    const float* __restrict__ W, const float* __restrict__ bias,
    const float* __restrict__ h, const float* __restrict__ extra,
    float* __restrict__ out0, float* __restrict__ out1, int OD, int mode) {
  __shared__ float hs[H_SZ];
  const int b = blockIdx.y;
  const int t = threadIdx.x;
  hs[t] = h[b * H_SZ + t];
  __syncthreads();
  const int j = blockIdx.x * 256 + t;
  float acc = bias[j];
  for (int k = 0; k < H_SZ; ++k) acc = fmaf(hs[k], W[k * OD + j], acc);
  if (mode == 0) {
    float x = fminf(fmaxf(acc, -3.0f), 3.0f);
    out0[b * OD + j] = x;
    out1[b * OD + j] = expf(-x);
  } else if (mode == 1) {
    out0[b * OD + j] = extra[b * OD + j] + 0.1f * acc;
  } else if (mode == 2) {
    out0[b * OD + j] = acc;
  } else {
    out0[b * OD + j] = 0.1f * acc;
  }
}

// ---------------------------------------------------------------------------
// per-batch: logd_sum, M = I + L^T D^-1 L, M^-1 (Gauss-Jordan, SPD), logdet
// ---------------------------------------------------------------------------
__global__ __launch_bounds__(256) void prep_kernel(
    const float* __restrict__ log_d, const float* __restrict__ d_inv,
    const float* __restrict__ L, float* __restrict__ Minv,
    float* __restrict__ log_det) {
  __shared__ float red[256];
  __shared__ float A[R_SZ][R_SZ + 1];
  __shared__ float In[R_SZ][R_SZ + 1];
  const int b = blockIdx.x;
  const int t = threadIdx.x;
  const int r = t >> 4, c = t & 15;

  // phase 1: sum of log_d (fixed-order tree reduction)
  float s = 0.0f;
  for (int d = t; d < D_SZ; d += 256) s += log_d[b * D_SZ + d];
  red[t] = s;
  __syncthreads();
  for (int off = 128; off > 0; off >>= 1) {
    if (t < off) red[t] += red[t + off];
    __syncthreads();
  }
  float ldacc = red[0];

  // phase 2: build M
  const float* Lb = L + b * (D_SZ * R_SZ);
  const float* db = d_inv + b * D_SZ;
  float acc = (r == c) ? 1.0f : 0.0f;
  for (int d = 0; d < D_SZ; ++d) {
    float di = db[d];
    acc = fmaf(Lb[d * R_SZ + r] * di, Lb[d * R_SZ + c], acc);
  }
  A[r][c] = acc;
  In[r][c] = (r == c) ? 1.0f : 0.0f;
  __syncthreads();

  // phase 3: Gauss-Jordan without pivoting (M is SPD)
  for (int k = 0; k < R_SZ; ++k) {
    float p = A[k][k];
    __syncthreads();
    ldacc += logf(p);
    float pinv = 1.0f / p;
    if (t < R_SZ) { A[k][t] *= pinv; In[k][t] *= pinv; }
    __syncthreads();
    float f = A[r][k];
    float akc = A[k][c];
    float ikc = In[k][c];
    __syncthreads();
    if (r != k) {
      A[r][c]  = fmaf(-f, akc, A[r][c]);
      In[r][c] = fmaf(-f, ikc, In[r][c]);
    }
    __syncthreads();
  }
  Minv[b * 256 + t] = In[r][c];
  if (t == 0) log_det[b] = ldacc;
}

// ---------------------------------------------------------------------------
// main kernel: one block per (batch, 16-row n-tile). 256 threads = 8 waves.
// ---------------------------------------------------------------------------
__global__ __launch_bounds__(256) void skew_main_kernel(
    const float* __restrict__ ctx, const float* __restrict__ mu,
    const float* __restrict__ dinv, const float* __restrict__ alpha,
    const float* __restrict__ L, const float* __restrict__ Minv,
    const float* __restrict__ log_det, float* __restrict__ out) {
  __shared__ float diff_s[TILE_N * PITCH];   // 33 KB
  __shared__ float mu_s[D_SZ];
  __shared__ float dinv_s[D_SZ];
  __shared__ float alpha_s[D_SZ];
  __shared__ float minv_s[R_SZ * R_SZ];
  __shared__ float wpart[8][R_SZ * R_SZ];    // per-wave partial w
  __shared__ float wfin[R_SZ][R_SZ];
  __shared__ float ctmp[R_SZ][R_SZ];
  __shared__ float mahal_s[TILE_N];
  __shared__ float skew_s[TILE_N];

  const int t    = threadIdx.x;
  const int wv   = t >> 5;
  const int lane = t & 31;
  const int b    = blockIdx.x >> 8;     // 256 tiles per batch
  const int tile = blockIdx.x & 255;
  const int n0   = tile * TILE_N;

  const float* mu_b    = mu    + b * D_SZ;
  const float* dinv_b  = dinv  + b * D_SZ;
  const float* alpha_b = alpha + b * D_SZ;
  const float* L_b     = L     + b * (D_SZ * R_SZ);

  for (int i = t; i < D_SZ; i += 256) {
    mu_s[i]    = mu_b[i];
    dinv_s[i]  = dinv_b[i];
    alpha_s[i] = alpha_b[i];
  }
  minv_s[t] = Minv[b * 256 + t];
  __syncthreads();

  v8f acc = {0.f, 0.f, 0.f, 0.f, 0.f, 0.f, 0.f, 0.f};
  float mahalA = 0.f, skewA = 0.f, mahalB = 0.f, skewB = 0.f;
  const int rowA = 2 * wv, rowB = 2 * wv + 1;
  const float* ctxA = ctx + ((size_t)(b * N_SZ + n0 + rowA)) * D_SZ;
  const float* ctxB = ctxA + D_SZ;

  for (int c0 = 0; c0 < D_SZ; c0 += CHUNK) {
    __syncthreads();  // previous chunk's LDS readers done
    // ---- load phase: coalesced float4, build diff, mahal/skew partials ----
    for (int it = 0; it < CHUNK / 128; ++it) {
      const int dl = it * 128 + 4 * lane;
      const int d  = c0 + dl;
      float4 xA = *(const float4*)(ctxA + d);
      float4 xB = *(const float4*)(ctxB + d);
      float m0 = mu_s[d], m1 = mu_s[d+1], m2 = mu_s[d+2], m3 = mu_s[d+3];
      float i0 = dinv_s[d], i1 = dinv_s[d+1], i2 = dinv_s[d+2], i3 = dinv_s[d+3];
      float a0 = alpha_s[d], a1 = alpha_s[d+1], a2 = alpha_s[d+2], a3 = alpha_s[d+3];
      float dA0 = xA.x - m0, dA1 = xA.y - m1, dA2 = xA.z - m2, dA3 = xA.w - m3;
      float dB0 = xB.x - m0, dB1 = xB.y - m1, dB2 = xB.z - m2, dB3 = xB.w - m3;
      mahalA = fmaf(dA0 * dA0, i0, mahalA); mahalA = fmaf(dA1 * dA1, i1, mahalA);
      mahalA = fmaf(dA2 * dA2, i2, mahalA); mahalA = fmaf(dA3 * dA3, i3, mahalA);
      mahalB = fmaf(dB0 * dB0, i0, mahalB); mahalB = fmaf(dB1 * dB1, i1, mahalB);
      mahalB = fmaf(dB2 * dB2, i2, mahalB); mahalB = fmaf(dB3 * dB3, i3, mahalB);
      skewA = fmaf(a0, dA0, skewA); skewA = fmaf(a1, dA1, skewA);
      skewA = fmaf(a2, dA2, skewA); skewA = fmaf(a3, dA3, skewA);
      skewB = fmaf(a0, dB0, skewB); skewB = fmaf(a1, dB1, skewB);
      skewB = fmaf(a2, dB2, skewB); skewB = fmaf(a3, dB3, skewB);
      *(float4*)&diff_s[rowA * PITCH + dl] = make_float4(dA0, dA1, dA2, dA3);
      *(float4*)&diff_s[rowB * PITCH + dl] = make_float4(dB0, dB1, dB2, dB3);
    }
    __syncthreads();
    // ---- WMMA phase: wave wv owns d-range [c0 + wv*64, +64), K=4 steps ----
    const int klo = (lane < 16) ? 0 : 2;
    const int nn  = lane & 15;   // B: column index (n), A: row index (r)
    const int dbase = c0 + wv * 64;
    for (int s = 0; s < 16; ++s) {
      const int d  = dbase + s * 4 + klo;
      const int dl = d - c0;
      v2f a;  // A = L^T chunk, 16(r) x 4(d): lane=r, VGPR0=K klo, VGPR1=K klo+1
      a.x = L_b[d * R_SZ + nn];
      a.y = L_b[(d + 1) * R_SZ + nn];
      v2f bb; // B = v chunk, 4(d) x 16(n): lane=n, VGPR0=K klo, VGPR1=K klo+1
      bb.x = diff_s[nn * PITCH + dl]     * dinv_s[d];
      bb.y = diff_s[nn * PITCH + dl + 1] * dinv_s[d + 1];
      acc = __builtin_amdgcn_wmma_f32_16x16x4_f32(
          false, a, false, bb, (short)0, acc, false, false);
    }
  }

  // ---- per-row reductions (rows rowA,rowB owned entirely by wave wv) ----
  for (int off = 16; off > 0; off >>= 1) {
    mahalA += __shfl_xor(mahalA, off, 32);
    mahalB += __shfl_xor(mahalB, off, 32);
    skewA  += __shfl_xor(skewA,  off, 32);
    skewB  += __shfl_xor(skewB,  off, 32);
  }
  if (lane == 0) {
    mahal_s[rowA] = mahalA; mahal_s[rowB] = mahalB;
    skew_s[rowA]  = skewA;  skew_s[rowB]  = skewB;
  }

  // ---- dump per-wave partial w (C/D layout: vgpr i, lane l ->
  //      r = i + (l<16?0:8), n = l&15) ----
  {
    const int nn = lane & 15;
    const int rbase = (lane < 16) ? 0 : 8;
#pragma unroll
    for (int i = 0; i < 8; ++i)
      wpart[wv][(rbase + i) * R_SZ + nn] = acc[i];
  }
  __syncthreads();

  // ---- deterministic cross-wave reduce of w ----
  const int r = t >> 4, n = t & 15;
  {
    float wsum = 0.f;
#pragma unroll
    for (int w = 0; w < 8; ++w) wsum += wpart[w][t];
    wfin[r][n] = wsum;
  }
  __syncthreads();

  // ---- correction: ctmp[r][n] = w[r][n] * sum_s Minv[r][s] w[s][n] ----
  {
    float tv = 0.f;
#pragma unroll
    for (int s2 = 0; s2 < R_SZ; ++s2)
      tv = fmaf(minv_s[r * R_SZ + s2], wfin[s2][n], tv);
    ctmp[r][n] = wfin[r][n] * tv;
  }
  __syncthreads();

  if (t < TILE_N) {
    float corr = 0.f;
#pragma unroll
    for (int rr = 0; rr < R_SZ; ++rr) corr += ctmp[rr][t];
    float m  = mahal_s[t];
    float sk = skew_s[t];
    float score = -0.5f * (m - corr + log_det[b]);
    float lsig  = fminf(sk, 0.0f) - log1pf(expf(-fabsf(sk)));
    out[b * N_SZ + n0 + t] = score + lsig;
  }
}

// ---------------------------------------------------------------------------
extern "C" void kernel_launch(void* const* d_in, const int* in_sizes, int n_in,
                              void* d_out, int out_size, void* d_ws, size_t ws_size,
                              hipStream_t stream) {
  const float* q    = (const float*)d_in[0];
  const float* ctx  = (const float*)d_in[1];
  const float* W1   = (const float*)d_in[2];
  const float* b1   = (const float*)d_in[3];
  const float* Wd   = (const float*)d_in[4];
  const float* bd   = (const float*)d_in[5];
  const float* WL   = (const float*)d_in[6];
  const float* bL   = (const float*)d_in[7];
  const float* Wdel = (const float*)d_in[8];
  const float* bdel = (const float*)d_in[9];
  const float* Wa   = (const float*)d_in[10];
  const float* ba   = (const float*)d_in[11];
  float* out = (float*)d_out;

  float* ws     = (float*)d_ws;
  float* h      = ws;                 // 8*256      = 2048
  float* logd   = h      + 2048;      // 8*1024     = 8192
  float* dinv   = logd   + 8192;
  float* muv    = dinv   + 8192;
  float* alpha  = muv    + 8192;
  float* Lw     = alpha  + 8192;      // 8*1024*16  = 131072
  float* Minv   = Lw     + 131072;    // 8*256      = 2048
  float* logdet = Minv   + 2048;      // 8

  backbone_kernel<<<dim3(B_SZ), dim3(256), 0, stream>>>(q, W1, b1, h);
  head_kernel<<<dim3(D_SZ / 256, B_SZ), dim3(256), 0, stream>>>(
      Wd, bd, h, nullptr, logd, dinv, D_SZ, 0);
  head_kernel<<<dim3(D_SZ / 256, B_SZ), dim3(256), 0, stream>>>(
      Wdel, bdel, h, q, muv, nullptr, D_SZ, 1);
  head_kernel<<<dim3(D_SZ / 256, B_SZ), dim3(256), 0, stream>>>(
      Wa, ba, h, nullptr, alpha, nullptr, D_SZ, 2);
  head_kernel<<<dim3((D_SZ * R_SZ) / 256, B_SZ), dim3(256), 0, stream>>>(
      WL, bL, h, nullptr, Lw, nullptr, D_SZ * R_SZ, 3);
  prep_kernel<<<dim3(B_SZ), dim3(256), 0, stream>>>(logd, dinv, Lw, Minv, logdet);
  skew_main_kernel<<<dim3(B_SZ * (N_SZ / TILE_N)), dim3(256), 0, stream>>>(
      ctx, muv, dinv, alpha, Lw, Minv, logdet, out);
}